// VQC_31593779429880
// MI455X (gfx1250) — compile-verified
//
#include <hip/hip_runtime.h>

// VQC statevector simulator for MI455X (gfx1250, wave32).
// One wave == one sample; 4096-float state lives in LDS (2x16KB ping-pong,
// 64KB per 2-wave block -> 5 blocks / 10 waves per 320KB WGP).
// Shared RY layers become three 16x16 Kronecker matrices applied with
// V_WMMA_F32_16X16X4_F32 (K=16 as 4 chained k=4 steps, two independent
// column-tile accumulator chains in flight for XDL latency hiding).
// CNOT ladder is a GF(2)-linear index permutation fused into the group-0
// operand gather: perm(base ^ c0) = perm(base) ^ perm(c0).

#define NQ      12
#define DIM     4096
#define DEPTH   8
#define NCLASS  10
#define WAVES_PER_BLOCK 2
#define THREADS (WAVES_PER_BLOCK * 32)

typedef __attribute__((ext_vector_type(2))) float v2f;
typedef __attribute__((ext_vector_type(8))) float v8f;

// Gather index for one layer's CNOT ladder (even pairs applied, then odd).
// new[x] = old[perm(x)]; composite evaluated in reverse gate order.
// This map is linear over GF(2): perm(x^y) == perm(x)^perm(y).
__device__ __forceinline__ int cnot_perm(int x) {
  x ^= ((x >> 10) & 1) << 9;   // CNOT(1,2)   wire q <-> bit (11-q)
  x ^= ((x >> 8)  & 1) << 7;   // CNOT(3,4)
  x ^= ((x >> 6)  & 1) << 5;   // CNOT(5,6)
  x ^= ((x >> 4)  & 1) << 3;   // CNOT(7,8)
  x ^= ((x >> 2)  & 1) << 1;   // CNOT(9,10)
  x ^= ((x >> 11) & 1) << 10;  // CNOT(0,1)
  x ^= ((x >> 9)  & 1) << 8;   // CNOT(2,3)
  x ^= ((x >> 7)  & 1) << 6;   // CNOT(4,5)
  x ^= ((x >> 5)  & 1) << 4;   // CNOT(6,7)
  x ^= ((x >> 3)  & 1) << 2;   // CNOT(8,9)
  x ^= ((x >> 1)  & 1) << 0;   // CNOT(10,11)
  return x;
}

// State viewed as S[16 contracted rows, 256 cols] per qubit group.
// G=0: wires 0-3 (idx bits 11-8), G=1: wires 4-7 (bits 7-4), G=2: wires 8-11 (bits 3-0).
// smap(r, c0+m) == smap(r, m) + coff(c0) with coff = c0 (G==0) or c0*16 (G==1,2),
// and the c0 field is bit-disjoint from the (r,m) fields.
template<int G>
__device__ __forceinline__ int smap(int r, int c) {
  if constexpr (G == 0)      return r * 256 + c;
  else if constexpr (G == 1) return (c >> 4) * 256 + r * 16 + (c & 15);
  else                       return c * 16 + r;
}

template<int G>
__device__ __forceinline__ int coff(int c0) {
  return (G == 0) ? c0 : (c0 << 4);
}

// U_g[i][j] = prod over 4 wires of RY 2x2 element (Kronecker product).
template<int G>
__device__ __forceinline__ float kron16(int i, int j, const float* wc, const float* ws) {
  float p = 1.0f;
  #pragma unroll
  for (int t = 0; t < 4; ++t) {
    const int q  = 4 * G + t;
    const int ib = (i >> (3 - t)) & 1;
    const int jb = (j >> (3 - t)) & 1;
    p *= (ib == jb) ? wc[q] : (ib ? ws[q] : -ws[q]);
  }
  return p;
}

// D[16,256] = U_g x S ; two 16-col tiles in flight, each K=16 as 4 WMMA steps.
// PERM fuses the CNOT-ladder permutation into the B-operand gather (src!=dst then).
template<int G, bool PERM>
__device__ __forceinline__ void apply_group(const float* src, float* dst,
                                            const float* wc, const float* ws,
                                            int lane) {
  const int m = lane & 15;   // N-lane / M-row slot
  const int h = lane >> 4;   // half-wave select (ISA 16x4 A / 4x16 B layouts)

  // A operand (U_g) built in-register once, reused across all 16 N-tiles.
  // ISA 32-bit A 16x4 layout: lanes 0-15 K={kk,kk+1}, lanes 16-31 K={kk+2,kk+3}.
  v2f a[4];
  #pragma unroll
  for (int kc = 0; kc < 4; ++kc) {
    const int j0 = 4 * kc + 2 * h;
    a[kc][0] = kron16<G>(m, j0,     wc, ws);
    a[kc][1] = kron16<G>(m, j0 + 1, wc, ws);
  }

  // Tile-invariant B-operand row bases (pre-permuted for PERM: linearity of
  // cnot_perm over the bit-disjoint c0 field makes per-tile work a single XOR).
  int b0[4], b1[4];
  #pragma unroll
  for (int kc = 0; kc < 4; ++kc) {
    const int r0 = 4 * kc + h;
    const int r1 = 4 * kc + 2 + h;
    b0[kc] = smap<G>(r0, m);
    b1[kc] = smap<G>(r1, m);
    if (PERM) { b0[kc] = cnot_perm(b0[kc]); b1[kc] = cnot_perm(b1[kc]); }
  }
  // Tile-invariant D row bases.
  int db[8];
  #pragma unroll
  for (int r = 0; r < 8; ++r) db[r] = smap<G>(r + 8 * h, m);

  #pragma unroll 1
  for (int c0 = 0; c0 < 256; c0 += 32) {
    const int c1 = c0 + 16;
    // Source tile offsets: wave-uniform (SALU) -- perm(c0) for the fused CNOT case.
    const int sA = PERM ? cnot_perm(c0) : coff<G>(c0);
    const int sB = PERM ? cnot_perm(c1) : coff<G>(c1);

    // B operands 4x16: VGPR0 rows {kk,kk+1}, VGPR1 rows {kk+2,kk+3} across halves.
    v2f bA[4], bB[4];
    #pragma unroll
    for (int kc = 0; kc < 4; ++kc) {
      if (PERM) {
        bA[kc][0] = src[b0[kc] ^ sA]; bA[kc][1] = src[b1[kc] ^ sA];
        bB[kc][0] = src[b0[kc] ^ sB]; bB[kc][1] = src[b1[kc] ^ sB];
      } else {
        bA[kc][0] = src[b0[kc] + sA]; bA[kc][1] = src[b1[kc] + sA];
        bB[kc][0] = src[b0[kc] + sB]; bB[kc][1] = src[b1[kc] + sB];
      }
    }

    // Two independent accumulator chains (hides WMMA RAW latency).
    v8f accA = {}, accB = {};
    #pragma unroll
    for (int kc = 0; kc < 4; ++kc) {
      accA = __builtin_amdgcn_wmma_f32_16x16x4_f32(
          false, a[kc], false, bA[kc], (short)0, accA, false, false);
      accB = __builtin_amdgcn_wmma_f32_16x16x4_f32(
          false, a[kc], false, bB[kc], (short)0, accB, false, false);
    }

    // D layout: lane L vgpr r -> D[r + 8*(L>>4), L&15]; linear dst offsets.
    const int dA = coff<G>(c0);
    const int dB = coff<G>(c1);
    #pragma unroll
    for (int r = 0; r < 8; ++r) {
      dst[db[r] + dA] = accA[r];
      dst[db[r] + dB] = accB[r];
    }
  }
  // LDS phase fence (in-place tiles / cross-group RAW through LDS).
  asm volatile("s_wait_dscnt 0" ::: "memory");
}

extern __shared__ float lds[];

__global__ void __launch_bounds__(THREADS, 1)
vqc_kernel(const float* __restrict__ X, const float* __restrict__ W,
           float* __restrict__ out, int batch) {
  const int lane = threadIdx.x & 31;
  const int wave = threadIdx.x >> 5;
  const int s    = blockIdx.x * WAVES_PER_BLOCK + wave;
  if (s >= batch) return;                 // wave-uniform; EXEC stays all-1

  float* buf0 = lds + wave * (2 * DIM);
  float* buf1 = buf0 + DIM;

  // ---- data-encoding layer: RY on |+>^12 is a closed-form product state ----
  float e0[NQ], e1[NQ];
  #pragma unroll
  for (int q = 0; q < NQ; ++q) {
    const float t = 0.5f * X[s * NQ + q];
    const float c = __cosf(t), sn = __sinf(t);
    e0[q] = c - sn;                       // amplitude factor, bit=0
    e1[q] = c + sn;                       // amplitude factor, bit=1
  }
  e0[0] *= 0.015625f; e1[0] *= 0.015625f; // fold 2^(-NQ/2) = 1/64 once

  float lo = 1.0f;                        // idx bits 4..0 = lane -> wires 7..11
  #pragma unroll
  for (int q = 7; q < 12; ++q)
    lo *= ((lane >> (11 - q)) & 1) ? e1[q] : e0[q];

  #pragma unroll 4
  for (int i = 0; i < 128; ++i) {         // idx bits 11..5 = i -> wires 0..6
    float hi = 1.0f;
    #pragma unroll
    for (int q = 0; q < 7; ++q)
      hi *= ((i >> (6 - q)) & 1) ? e1[q] : e0[q];
    buf0[i * 32 + lane] = hi * lo;
  }
  asm volatile("s_wait_dscnt 0" ::: "memory");

  // ---- 8 variational layers: CNOT perm (fused) + 3 Kronecker WMMA passes ----
  float* cur = buf0;
  float* alt = buf1;
  #pragma unroll 1
  for (int k = 0; k < DEPTH; ++k) {
    float wc[NQ], ws[NQ];
    #pragma unroll
    for (int q = 0; q < NQ; ++q) {
      const float t = 0.5f * W[k * NQ + q];
      wc[q] = __cosf(t);
      ws[q] = __sinf(t);
    }
    apply_group<0, true >(cur, alt, wc, ws, lane);  // gather through cnot_perm
    apply_group<1, false>(alt, alt, wc, ws, lane);  // in place
    apply_group<2, false>(alt, alt, wc, ws, lane);  // in place
    float* t = cur; cur = alt; alt = t;
  }

  // ---- <Z_p> = sum probs with sign of bit (11-p) ----
  float acc[NCLASS];
  #pragma unroll
  for (int p = 0; p < NCLASS; ++p) acc[p] = 0.0f;
  #pragma unroll 4
  for (int i = 0; i < 128; ++i) {
    const int idx = i * 32 + lane;
    const float a  = cur[idx];
    const float p2 = a * a;
    #pragma unroll
    for (int p = 0; p < NCLASS; ++p)
      acc[p] += ((idx >> (11 - p)) & 1) ? -p2 : p2;
  }
  #pragma unroll
  for (int p = 0; p < NCLASS; ++p)
    #pragma unroll
    for (int off = 16; off >= 1; off >>= 1)
      acc[p] += __shfl_xor(acc[p], off, 32);

  if (lane == 0) {
    #pragma unroll
    for (int p = 0; p < NCLASS; ++p)
      out[s * NCLASS + p] = acc[p];
  }
}

extern "C" void kernel_launch(void* const* d_in, const int* in_sizes, int n_in,
                              void* d_out, int out_size, void* d_ws, size_t ws_size,
                              hipStream_t stream) {
  (void)n_in; (void)out_size; (void)d_ws; (void)ws_size;
  const float* X = (const float*)d_in[0];   // [BATCH, NQ] fp32
  const float* W = (const float*)d_in[1];   // [DEPTH, NQ] fp32
  float* out = (float*)d_out;               // [BATCH, NCLASS] fp32

  const int batch  = in_sizes[0] / NQ;
  const int blocks = (batch + WAVES_PER_BLOCK - 1) / WAVES_PER_BLOCK;
  const size_t lds_bytes = (size_t)WAVES_PER_BLOCK * 2 * DIM * sizeof(float); // 64 KB

  vqc_kernel<<<blocks, THREADS, lds_bytes, stream>>>(X, W, out, batch);
}